// PixelPrototypeDistanceLoss_22436909154962
// MI455X (gfx1250) — compile-verified
//
#include <hip/hip_runtime.h>

typedef float v2f __attribute__((ext_vector_type(2)));
typedef float v8f __attribute__((ext_vector_type(8)));

#define IGNORE_INDEX 19
#define Bn   8
#define Dn   256
#define HWn  16384           // 128*128
#define Nn   (Bn * HWn)      // 131072 pixels
#define Cn   19
#define PROTO_STRIDE 260     // 256 + 4 pad -> classes map to distinct LDS banks
#define SIM_STRIDE   34      // 32 classes + 2 pad -> disjoint bank ranges for the 2 half-waves
#define PIX_PER_WAVE  16
#define WAVES_PER_BLK 8
#define PIX_PER_BLK   (PIX_PER_WAVE * WAVES_PER_BLK)   // 128
#define NBLOCKS       (Nn / PIX_PER_BLK)               // 1024

// ---------------------------------------------------------------------------
// Main kernel: one wave computes a 16-pixel x 32-class sim tile with f32 WMMA,
// gathers the true-class logit per pixel, accumulates (1-logit)^2 over valid
// pixels into per-block partials.
// ---------------------------------------------------------------------------
__global__ __launch_bounds__(256) void ppd_main(const float* __restrict__ emb,
                                                const int*   __restrict__ lb,
                                                const float* __restrict__ sq,
                                                float*       __restrict__ partial)
{
    __shared__ float protoLds[32 * PROTO_STRIDE];                    // 33.3 KB (rows 19..31 zero)
    __shared__ float simLds[WAVES_PER_BLK * PIX_PER_WAVE * SIM_STRIDE]; // 17.4 KB
    __shared__ float redS[WAVES_PER_BLK];
    __shared__ float redC[WAVES_PER_BLK];

    const int tid = threadIdx.x;

    // Load prototype table into LDS, zero-padding classes 19..31 and cols 256..259.
    for (int i = tid; i < 32 * PROTO_STRIDE; i += 256) {
        int c = i / PROTO_STRIDE;
        int k = i - c * PROTO_STRIDE;
        float v = 0.0f;
        if (c < Cn && k < Dn) v = sq[c * Dn + k];
        protoLds[i] = v;
    }
    __syncthreads();

    const int lane  = tid & 31;
    const int wave  = tid >> 5;
    const int m     = lane & 15;     // row (pixel) within tile / class column for B
    const int half  = lane >> 4;     // 0: K0,K1  1: K2,K3   (documented A layout)
    const int kHalf = half * 2;

    const int pix0 = blockIdx.x * PIX_PER_BLK + wave * PIX_PER_WAVE; // 16-aligned, inside one b-plane
    const int b    = pix0 >> 14;          // / HWn
    const int hw0  = pix0 & (HWn - 1);

    // A element stream: emb[b, k, hw0+m], k advances 4 planes per iteration
    const float* aPtr = emb + ((size_t)b * Dn + kHalf) * (size_t)HWn + hw0 + m;
    // B columns from LDS (same K->(half,vgpr) mapping as A, so layout-permutation-safe)
    const float* pB0 = &protoLds[m * PROTO_STRIDE + kHalf];          // classes 0..15
    const float* pB1 = &protoLds[(16 + m) * PROTO_STRIDE + kHalf];   // classes 16..31 (19..31 zero)

    v8f acc0 = {};
    v8f acc1 = {};

#pragma unroll 4
    for (int k0 = 0; k0 < Dn; k0 += 4) {
        v2f a;
        a.x = aPtr[0];
        a.y = aPtr[HWn];
        v2f bv0;
        bv0.x = pB0[k0];
        bv0.y = pB0[k0 + 1];
        v2f bv1;
        bv1.x = pB1[k0];
        bv1.y = pB1[k0 + 1];
        acc0 = __builtin_amdgcn_wmma_f32_16x16x4_f32(false, a, false, bv0,
                                                     (short)0, acc0, false, false);
        acc1 = __builtin_amdgcn_wmma_f32_16x16x4_f32(false, a, false, bv1,
                                                     (short)0, acc1, false, false);
        aPtr += 4 * HWn;
    }

    // Spill sim tile to LDS: D layout = VGPR r holds rows r (lanes 0-15) and r+8 (lanes 16-31), N=lane%16
    float* sw = &simLds[wave * PIX_PER_WAVE * SIM_STRIDE];
    const int mBase = half * 8;
#pragma unroll
    for (int r = 0; r < 8; ++r) {
        sw[(mBase + r) * SIM_STRIDE + m]        = acc0[r];
        sw[(mBase + r) * SIM_STRIDE + 16 + m]   = acc1[r];
    }
    __syncthreads();

    // Gather true-class logit per pixel (lanes 0-15 own the 16 pixels)
    float ssq = 0.0f, cnt = 0.0f;
    if (half == 0) {
        int label = lb[pix0 + m];
        if (label != IGNORE_INDEX) {
            float logit = sw[m * SIM_STRIDE + label];
            float d = 1.0f - logit;
            ssq = d * d;
            cnt = 1.0f;
        }
    }

    // wave32 butterfly reduction
#pragma unroll
    for (int off = 16; off > 0; off >>= 1) {
        ssq += __shfl_xor(ssq, off);
        cnt += __shfl_xor(cnt, off);
    }
    if (lane == 0) { redS[wave] = ssq; redC[wave] = cnt; }
    __syncthreads();

    if (tid == 0) {
        float s = 0.0f, c = 0.0f;
#pragma unroll
        for (int w = 0; w < WAVES_PER_BLK; ++w) { s += redS[w]; c += redC[w]; }
        partial[2 * blockIdx.x + 0] = s;
        partial[2 * blockIdx.x + 1] = c;
    }
}

// ---------------------------------------------------------------------------
// Deterministic final reduction (fixed order, no float atomics).
// ---------------------------------------------------------------------------
__global__ __launch_bounds__(256) void ppd_reduce(const float* __restrict__ partial,
                                                  float*       __restrict__ out)
{
    __shared__ float sS[256];
    __shared__ float sC[256];
    const int tid = threadIdx.x;
    float s = 0.0f, c = 0.0f;
    for (int i = tid; i < NBLOCKS; i += 256) {
        s += partial[2 * i + 0];
        c += partial[2 * i + 1];
    }
    sS[tid] = s;
    sC[tid] = c;
    __syncthreads();
    for (int st = 128; st > 0; st >>= 1) {
        if (tid < st) {
            sS[tid] += sS[tid + st];
            sC[tid] += sC[tid + st];
        }
        __syncthreads();
    }
    if (tid == 0) out[0] = sS[0] / sC[0];
}

extern "C" void kernel_launch(void* const* d_in, const int* in_sizes, int n_in,
                              void* d_out, int out_size, void* d_ws, size_t ws_size,
                              hipStream_t stream)
{
    const float* emb = (const float*)d_in[0];   // [8, 256, 128, 128] f32
    const int*   lb  = (const int*)d_in[1];     // [8, 128, 128] int
    const float* sq  = (const float*)d_in[2];   // [19, 256] f32
    float* partial   = (float*)d_ws;            // 1024 * 2 floats
    float* out       = (float*)d_out;

    ppd_main<<<NBLOCKS, 256, 0, stream>>>(emb, lb, sq, partial);
    ppd_reduce<<<1, 256, 0, stream>>>(partial, out);
}